// Model_11862699672212
// MI455X (gfx1250) — compile-verified
//
#include <hip/hip_runtime.h>

// Fused deformable-conv kernel for MI455X (gfx1250, wave32).
// Stage 1 (per wave): offset conv (pixels x 27) @ (27 x 18) via
//   v_wmma_f32_16x16x4_f32, K padded 27->28 (7 steps), N padded 18->32 (2 tiles),
//   two M=16 pixel tiles -> 28 WMMA ops per wave, fp32 throughout.
// Stage 2: per-lane bilinear gather (L2-resident x) + 27->3 projection.

typedef __attribute__((ext_vector_type(2))) float v2f;
typedef __attribute__((ext_vector_type(8))) float v8f;

#define Bn   16
#define Cn   3
#define Hn   384
#define Wn   384
#define HoN  382
#define WoN  382
#define O18  18
#define K27  27

// im2col element for the 3x3 VALID conv at output pixel (ho, wo), k = c*9 + r*3 + s.
// Branchless: clamp k and column, select 0 for pad lanes (keeps EXEC full for WMMA).
__device__ __forceinline__ float patch_at(const float* __restrict__ x,
                                          int b, int ho, int wo, int k) {
  const int kc  = k < (K27 - 1) ? k : (K27 - 1);
  const int c   = kc / 9;
  const int rem = kc - c * 9;
  const int rr  = rem / 3;
  const int ss  = rem - rr * 3;
  int col = wo + ss;  col = col < (Wn - 1) ? col : (Wn - 1);
  const float v = x[((b * Cn + c) * Hn + (ho + rr)) * Wn + col];
  return (k < K27) ? v : 0.0f;
}

// conv_w is (O=18, C=3, 3, 3); W1[k][n] with k = c*9 + r*3 + s.
__device__ __forceinline__ float w1_at(const float* __restrict__ cw, int k, int n) {
  const int kc = k < (K27 - 1) ? k : (K27 - 1);
  const int nc = n < (O18 - 1) ? n : (O18 - 1);
  const float v = cw[nc * K27 + kc];
  return (k < K27 && n < O18) ? v : 0.0f;
}

__global__ __launch_bounds__(128)
void dcn_fused_wmma(const float* __restrict__ x,
                    const float* __restrict__ conv_w,
                    const float* __restrict__ conv_b,
                    const float* __restrict__ dcn_w,
                    const float* __restrict__ dcn_b,
                    float* __restrict__ out) {
  __shared__ float soff[4][32][O18];   // per-wave: [pixel][offset-channel]

  const int tid   = threadIdx.x;
  const int wave  = tid >> 5;          // wave32
  const int lane  = tid & 31;
  const int b     = blockIdx.z;
  const int ho    = blockIdx.y;
  const int wbase = blockIdx.x * 128 + wave * 32;   // first pixel wo of this wave

  // ---- Stage 1: offset conv via WMMA f32 16x16x4 ----
  const int mA  = lane & 15;           // A row (pixel) / B column within tile
  const int kHi = (lane >> 4) << 1;    // lanes 0-15 -> K+{0,1}, lanes 16-31 -> K+{2,3}

  v8f acc00 = {}; v8f acc01 = {}; v8f acc10 = {}; v8f acc11 = {};

  #pragma unroll
  for (int kk = 0; kk < 7; ++kk) {     // K = 28 / 4
    const int k0 = kk * 4 + kHi;
    const int k1 = k0 + 1;

    v2f a0, a1, b0, b1;
    a0.x = patch_at(x, b, ho, wbase + mA,       k0);
    a0.y = patch_at(x, b, ho, wbase + mA,       k1);
    a1.x = patch_at(x, b, ho, wbase + 16 + mA,  k0);
    a1.y = patch_at(x, b, ho, wbase + 16 + mA,  k1);
    b0.x = w1_at(conv_w, k0, mA);
    b0.y = w1_at(conv_w, k1, mA);
    b1.x = w1_at(conv_w, k0, 16 + mA);
    b1.y = w1_at(conv_w, k1, 16 + mA);

    acc00 = __builtin_amdgcn_wmma_f32_16x16x4_f32(false, a0, false, b0, (short)0, acc00, false, false);
    acc01 = __builtin_amdgcn_wmma_f32_16x16x4_f32(false, a0, false, b1, (short)0, acc01, false, false);
    acc10 = __builtin_amdgcn_wmma_f32_16x16x4_f32(false, a1, false, b0, (short)0, acc10, false, false);
    acc11 = __builtin_amdgcn_wmma_f32_16x16x4_f32(false, a1, false, b1, (short)0, acc11, false, false);
  }

  // D layout: VGPR r, lane l -> row M = r + 8*(l>>4), col N = l&15.
  // Scatter offsets (+conv bias) to LDS so each lane can read all 18 for its pixel.
  {
    const int n0     = lane & 15;
    const int mBase  = (lane >> 4) << 3;
    const float bias0 = conv_b[n0];
    #pragma unroll
    for (int r = 0; r < 8; ++r) {
      const int pix = mBase + r;
      soff[wave][pix][n0]      = acc00[r] + bias0;
      soff[wave][16 + pix][n0] = acc10[r] + bias0;
    }
    if (n0 < O18 - 16) {               // columns 16,17 from the second N-tile
      const int n1 = 16 + n0;
      const float bias1 = conv_b[n1];
      #pragma unroll
      for (int r = 0; r < 8; ++r) {
        const int pix = mBase + r;
        soff[wave][pix][n1]      = acc01[r] + bias1;
        soff[wave][16 + pix][n1] = acc11[r] + bias1;
      }
    }
  }
  __syncthreads();

  // ---- Stage 2: bilinear sampling + 27->3 projection (one pixel per lane) ----
  const int wo = wbase + lane;
  if (wo < WoN) {
    float o0 = dcn_b[0], o1 = dcn_b[1], o2 = dcn_b[2];

    #pragma unroll
    for (int k = 0; k < 9; ++k) {
      const int ky = k / 3, kx = k % 3;
      const float py = soff[wave][lane][2 * k]     + (float)(ho + ky);
      const float px = soff[wave][lane][2 * k + 1] + (float)(wo + kx);
      const float fy = floorf(py);
      const float fx = floorf(px);
      const float wy = py - fy;
      const float wx = px - fx;
      const int y0 = (int)fy, x0 = (int)fx;
      const int y1 = y0 + 1,  x1 = x0 + 1;

      const bool vy0 = (y0 >= 0) & (y0 < Hn);
      const bool vy1 = (y1 >= 0) & (y1 < Hn);
      const bool vx0 = (x0 >= 0) & (x0 < Wn);
      const bool vx1 = (x1 >= 0) & (x1 < Wn);
      const int y0c = min(max(y0, 0), Hn - 1);
      const int y1c = min(max(y1, 0), Hn - 1);
      const int x0c = min(max(x0, 0), Wn - 1);
      const int x1c = min(max(x1, 0), Wn - 1);

      const float w00 = (1.f - wy) * (1.f - wx) * ((vy0 && vx0) ? 1.f : 0.f);
      const float w01 = (1.f - wy) * wx         * ((vy0 && vx1) ? 1.f : 0.f);
      const float w10 = wy * (1.f - wx)         * ((vy1 && vx0) ? 1.f : 0.f);
      const float w11 = wy * wx                 * ((vy1 && vx1) ? 1.f : 0.f);

      #pragma unroll
      for (int c = 0; c < Cn; ++c) {
        const float* xc = x + (size_t)(b * Cn + c) * Hn * Wn;
        const float s = w00 * xc[y0c * Wn + x0c]
                      + w01 * xc[y0c * Wn + x1c]
                      + w10 * xc[y1c * Wn + x0c]
                      + w11 * xc[y1c * Wn + x1c];
        o0 += s * dcn_w[(0 * Cn + c) * 9 + k];
        o1 += s * dcn_w[(1 * Cn + c) * 9 + k];
        o2 += s * dcn_w[(2 * Cn + c) * 9 + k];
      }
    }

    const size_t plane = (size_t)HoN * WoN;
    const size_t base  = ((size_t)b * 3 * HoN + ho) * WoN + wo;
    out[base]             = o0;
    out[base + plane]     = o1;
    out[base + 2 * plane] = o2;
  }
}

extern "C" void kernel_launch(void* const* d_in, const int* in_sizes, int n_in,
                              void* d_out, int out_size, void* d_ws, size_t ws_size,
                              hipStream_t stream) {
  (void)in_sizes; (void)n_in; (void)out_size; (void)d_ws; (void)ws_size;
  const float* x      = (const float*)d_in[0];
  const float* conv_w = (const float*)d_in[1];
  const float* conv_b = (const float*)d_in[2];
  const float* dcn_w  = (const float*)d_in[3];
  const float* dcn_b  = (const float*)d_in[4];
  float* out = (float*)d_out;

  dim3 grid((WoN + 127) / 128, HoN, Bn);   // (3, 382, 16)
  dcn_fused_wmma<<<grid, 128, 0, stream>>>(x, conv_w, conv_b, dcn_w, dcn_b, out);
}